// MambaEncoder_58007828299850
// MI455X (gfx1250) — compile-verified
//
#include <hip/hip_runtime.h>
#include <math.h>

// ---------------- problem constants (match reference) ----------------
constexpr int BATCH  = 8;
constexpr int SEQ    = 8192;
constexpr int DMODEL = 128;
constexpr int DINNER = 256;
constexpr int DSTATE = 16;
constexpr int DTRANK = 8;
constexpr int DEPTHN = 4;
constexpr int ROWS   = BATCH * SEQ;      // 65536 token rows

typedef __attribute__((ext_vector_type(2))) float v2f;
typedef __attribute__((ext_vector_type(8))) float v8f;

__device__ __forceinline__ float silu_f(float x) { return x / (1.f + __expf(-x)); }

// ---------------- 1) input projection (conv1d k=1 == pointwise) ----------------
__global__ void proj_in_kernel(const float* __restrict__ x, const float* __restrict__ pw,
                               const float* __restrict__ pb, float* __restrict__ h) {
    int id = blockIdx.x * 256 + threadIdx.x;        // id = bl*128 + d
    int d  = id & (DMODEL - 1);
    int bl = id >> 7;
    int b  = bl / SEQ, l = bl % SEQ;
    const float* xb = x + (size_t)b * 3 * SEQ + l;  // x[b, c, l], stride SEQ between c
    float acc = pb[d]
              + xb[0]       * pw[d * 3 + 0]
              + xb[SEQ]     * pw[d * 3 + 1]
              + xb[2 * SEQ] * pw[d * 3 + 2];
    h[(size_t)bl * DMODEL + d] = acc;
}

// ---------------- 2) layernorm over d_model=128 (one wave32 per row) ----------------
__global__ void layernorm_kernel(const float* __restrict__ x, const float* __restrict__ w,
                                 const float* __restrict__ b, float* __restrict__ out) {
    int wave = threadIdx.x >> 5;
    int lane = threadIdx.x & 31;
    size_t row = (size_t)blockIdx.x * 8 + wave;
    const float* xr = x + row * DMODEL;
    float v0 = xr[lane], v1 = xr[lane + 32], v2 = xr[lane + 64], v3 = xr[lane + 96];
    float s = v0 + v1 + v2 + v3;
    for (int m = 16; m >= 1; m >>= 1) s += __shfl_xor(s, m, 32);
    float mu = s * (1.f / DMODEL);
    float d0 = v0 - mu, d1 = v1 - mu, d2 = v2 - mu, d3 = v3 - mu;
    float q = d0 * d0 + d1 * d1 + d2 * d2 + d3 * d3;
    for (int m = 16; m >= 1; m >>= 1) q += __shfl_xor(q, m, 32);
    float rstd = rsqrtf(q * (1.f / DMODEL) + 1e-5f);
    float* o = out + row * DMODEL;
    o[lane]      = d0 * rstd * w[lane]      + b[lane];
    o[lane + 32] = d1 * rstd * w[lane + 32] + b[lane + 32];
    o[lane + 64] = d2 * rstd * w[lane + 64] + b[lane + 64];
    o[lane + 96] = d3 * rstd * w[lane + 96] + b[lane + 96];
}

// ---------------- 3) in_proj GEMM: xz[M,512] = xn[M,128] @ in_w[512,128]^T ----------------
// WMMA f32 16x16x4. Block: 256 thr = 8 waves; wave -> one 16x16 tile; block = 16 rows x 128 cols.
// grid = (512/128, ROWS/16). Epilogue splits cols [0,256)->u_raw, [256,512)->z.
__global__ void gemm_inproj_kernel(const float* __restrict__ A, const float* __restrict__ W,
                                   float* __restrict__ uraw, float* __restrict__ zb) {
    __shared__ float lA[16 * DMODEL];
    size_t row0 = (size_t)blockIdx.y * 16;
    const float* As = A + row0 * DMODEL;             // 16 contiguous rows
    for (int i = threadIdx.x; i < 16 * DMODEL; i += 256) lA[i] = As[i];
    __syncthreads();

    int wave = threadIdx.x >> 5, lane = threadIdx.x & 31;
    int half = lane >> 4, lm = lane & 15;
    int n0 = blockIdx.x * 128 + wave * 16;
    const float* Wr = W + (size_t)(n0 + lm) * DMODEL; // B frag: lane=N, vgpr/half=K
    const float* Ar = lA + lm * DMODEL;               // A frag: lane=M, vgpr/half=K
    v8f c = {0.f, 0.f, 0.f, 0.f, 0.f, 0.f, 0.f, 0.f};
    #pragma unroll 4
    for (int k = 0; k < DMODEL; k += 4) {
        v2f a, bb;
        a.x  = Ar[k + 2 * half];  a.y  = Ar[k + 2 * half + 1];
        bb.x = Wr[k + 2 * half];  bb.y = Wr[k + 2 * half + 1];
        c = __builtin_amdgcn_wmma_f32_16x16x4_f32(false, a, false, bb, (short)0, c, false, false);
    }
    int col = n0 + lm;
    #pragma unroll
    for (int r = 0; r < 8; r++) {
        size_t row = row0 + r + 8 * half;             // C/D layout: vgpr r -> M=r (lanes<16) / r+8
        float v = c[r];
        if (col < DINNER) uraw[row * DINNER + col] = v;
        else              zb[row * DINNER + (col - DINNER)] = v;
    }
}

// ---------------- 4) causal depthwise conv (k=4) + SiLU ----------------
__global__ void conv_silu_kernel(const float* __restrict__ uraw, const float* __restrict__ cw,
                                 const float* __restrict__ cb, float* __restrict__ u) {
    size_t id = (size_t)blockIdx.x * 256 + threadIdx.x;  // id = bl*256 + d
    int d = (int)(id & (DINNER - 1));
    size_t bl = id >> 8;
    long l = (long)(bl & (SEQ - 1));
    float acc = cb[d];
    const float* w4 = cw + d * 4;
    #pragma unroll
    for (int k = 0; k < 4; k++) {
        long lt = l + k - 3;
        if (lt >= 0) acc += uraw[(bl + (size_t)(k - 3)) * DINNER + d] * w4[k];
    }
    u[id] = silu_f(acc);
}

// ---------------- 5) x_proj (40 outs) + dt head + softplus, per token row ----------------
// block = 64 threads per row. dbl[0:8]->dt, [8:24]->B, [24:40]->C
__global__ void xproj_dt_kernel(const float* __restrict__ u, const float* __restrict__ xw,
                                const float* __restrict__ dw, const float* __restrict__ db,
                                float* __restrict__ dtf, float* __restrict__ Bm,
                                float* __restrict__ Cm) {
    __shared__ float ur[DINNER];
    __shared__ float dbl[DTRANK + 2 * DSTATE];
    size_t row = blockIdx.x;
    int t = threadIdx.x;
    const float* urow = u + row * DINNER;
    ((float4*)ur)[t] = ((const float4*)urow)[t];     // 64 * 4 = 256 floats
    __syncthreads();
    if (t < DTRANK + 2 * DSTATE) {
        const float* wr = xw + t * DINNER;
        float s = 0.f;
        for (int j = 0; j < DINNER; j++) s += ur[j] * wr[j];
        dbl[t] = s;
    }
    __syncthreads();
    #pragma unroll
    for (int j = 0; j < 4; j++) {
        int d = t * 4 + j;
        float s = db[d];
        #pragma unroll
        for (int r = 0; r < DTRANK; r++) s += dbl[r] * dw[d * DTRANK + r];
        dtf[row * DINNER + d] = (s > 20.f) ? s : log1pf(__expf(s));   // softplus
    }
    if (t < DSTATE)          Bm[row * DSTATE + t] = dbl[DTRANK + t];
    else if (t < 2 * DSTATE) Cm[row * DSTATE + (t - DSTATE)] = dbl[DTRANK + t];
}

// ---------------- 6) selective scan + output gate, state in registers ----------------
// grid = BATCH blocks of DINNER threads; thread = one (b,d) channel; h[16] in VGPRs.
// B/C rows are wave-uniform broadcast loads (L2 resident). Writes gated y over z in-place.
__global__ void scan_kernel(const float* __restrict__ dtf, const float* __restrict__ u,
                            float* __restrict__ zio, const float* __restrict__ Bm,
                            const float* __restrict__ Cm, const float* __restrict__ A_log,
                            const float* __restrict__ Dp) {
    int b = blockIdx.x, d = threadIdx.x;
    float A[DSTATE], h[DSTATE];
    #pragma unroll
    for (int s = 0; s < DSTATE; s++) { A[s] = -__expf(A_log[d * DSTATE + s]); h[s] = 0.f; }
    float Dv = Dp[d];
    size_t base = (size_t)b * SEQ;
    for (int t = 0; t < SEQ; t++) {
        size_t r = base + t;
        float dt = dtf[r * DINNER + d];
        float uu = u[r * DINNER + d];
        float zz = zio[r * DINNER + d];
        float du = dt * uu;
        const float* Bv = Bm + r * DSTATE;
        const float* Cv = Cm + r * DSTATE;
        float y = 0.f;
        #pragma unroll
        for (int s = 0; s < DSTATE; s++) {
            h[s] = h[s] * __expf(dt * A[s]) + du * Bv[s];   // h = h*exp(dt*A) + dt*u*B
            y   += h[s] * Cv[s];                            // y = h . C (post-update)
        }
        zio[r * DINNER + d] = (y + uu * Dv) * silu_f(zz);   // gate, in-place over z
    }
}

// ---------------- 7) out_proj GEMM (accumulate residual): h[M,128] += yg[M,256] @ out_w[128,256]^T
__global__ void gemm_out_kernel(const float* __restrict__ A, const float* __restrict__ W,
                                float* __restrict__ h) {
    __shared__ float lA[16 * DINNER];
    size_t row0 = (size_t)blockIdx.x * 16;
    const float* As = A + row0 * DINNER;
    for (int i = threadIdx.x; i < 16 * DINNER; i += 256) lA[i] = As[i];
    __syncthreads();

    int wave = threadIdx.x >> 5, lane = threadIdx.x & 31;
    int half = lane >> 4, lm = lane & 15;
    int n0 = wave * 16;                               // N=128 = 8 waves * 16
    const float* Wr = W + (size_t)(n0 + lm) * DINNER;
    const float* Ar = lA + lm * DINNER;
    v8f c = {0.f, 0.f, 0.f, 0.f, 0.f, 0.f, 0.f, 0.f};
    #pragma unroll 4
    for (int k = 0; k < DINNER; k += 4) {
        v2f a, bb;
        a.x  = Ar[k + 2 * half];  a.y  = Ar[k + 2 * half + 1];
        bb.x = Wr[k + 2 * half];  bb.y = Wr[k + 2 * half + 1];
        c = __builtin_amdgcn_wmma_f32_16x16x4_f32(false, a, false, bb, (short)0, c, false, false);
    }
    int col = n0 + lm;
    #pragma unroll
    for (int r = 0; r < 8; r++) {
        size_t row = row0 + r + 8 * half;
        h[row * DMODEL + col] += c[r];                // residual accumulate
    }
}

// ---------------- 8) mean over L of the final LN output (deterministic tree) ----------------
__global__ void mean_kernel(const float* __restrict__ hln, float* __restrict__ outg) {
    __shared__ float red[256];
    int b = blockIdx.x >> 7;
    int d = blockIdx.x & 127;
    const float* p = hln + (size_t)b * SEQ * DMODEL + d;
    float s = 0.f;
    for (int t = threadIdx.x; t < SEQ; t += 256) s += p[(size_t)t * DMODEL];
    red[threadIdx.x] = s;
    __syncthreads();
    for (int m = 128; m >= 1; m >>= 1) {
        if (threadIdx.x < m) red[threadIdx.x] += red[threadIdx.x + m];
        __syncthreads();
    }
    if (threadIdx.x == 0) outg[blockIdx.x] = red[0] * (1.f / SEQ);
}

// ---------------- host driver ----------------
extern "C" void kernel_launch(void* const* d_in, const int* in_sizes, int n_in,
                              void* d_out, int out_size, void* d_ws, size_t ws_size,
                              hipStream_t stream) {
    const float* x       = (const float*)d_in[0];
    const float* proj_w  = (const float*)d_in[1];
    const float* proj_b  = (const float*)d_in[2];
    const float* ln_w    = (const float*)d_in[3];
    const float* ln_b    = (const float*)d_in[4];
    const float* in_w    = (const float*)d_in[5];
    const float* conv_w  = (const float*)d_in[6];
    const float* conv_b  = (const float*)d_in[7];
    const float* xproj_w = (const float*)d_in[8];
    const float* dt_w    = (const float*)d_in[9];
    const float* dt_b    = (const float*)d_in[10];
    const float* A_log   = (const float*)d_in[11];
    const float* Dp      = (const float*)d_in[12];
    const float* out_w   = (const float*)d_in[13];
    const float* lnout_w = (const float*)d_in[14];
    const float* lnout_b = (const float*)d_in[15];

    // workspace layout (floats); buffers recycled across stages
    float* ws = (float*)d_ws;
    size_t off = 0;
    float* h    = ws + off; off += (size_t)ROWS * DMODEL;   // residual stream
    float* xn   = ws + off; off += (size_t)ROWS * DMODEL;   // layernormed h
    float* buf1 = ws + off; off += (size_t)ROWS * DINNER;   // u_raw, then dt_full
    float* buf2 = ws + off; off += (size_t)ROWS * DINNER;   // z, then gated y
    float* buf3 = ws + off; off += (size_t)ROWS * DINNER;   // u (post conv+silu)
    float* Bm   = ws + off; off += (size_t)ROWS * DSTATE;
    float* Cm   = ws + off; off += (size_t)ROWS * DSTATE;

    proj_in_kernel<<<ROWS * DMODEL / 256, 256, 0, stream>>>(x, proj_w, proj_b, h);

    for (int i = 0; i < DEPTHN; i++) {
        layernorm_kernel<<<ROWS / 8, 256, 0, stream>>>(h, ln_w + i * DMODEL, ln_b + i * DMODEL, xn);

        dim3 g1(2 * DINNER / 128, ROWS / 16);
        gemm_inproj_kernel<<<g1, 256, 0, stream>>>(xn, in_w + (size_t)i * 2 * DINNER * DMODEL,
                                                   buf1, buf2);

        conv_silu_kernel<<<ROWS * DINNER / 256, 256, 0, stream>>>(
            buf1, conv_w + (size_t)i * DINNER * 4, conv_b + (size_t)i * DINNER, buf3);

        xproj_dt_kernel<<<ROWS, 64, 0, stream>>>(
            buf3, xproj_w + (size_t)i * (DTRANK + 2 * DSTATE) * DINNER,
            dt_w + (size_t)i * DINNER * DTRANK, dt_b + (size_t)i * DINNER, buf1, Bm, Cm);

        scan_kernel<<<BATCH, DINNER, 0, stream>>>(
            buf1, buf3, buf2, Bm, Cm, A_log + (size_t)i * DINNER * DSTATE, Dp + (size_t)i * DINNER);

        gemm_out_kernel<<<ROWS / 16, 256, 0, stream>>>(
            buf2, out_w + (size_t)i * DMODEL * DINNER, h);
    }

    float* outp = (float*)d_out;
    layernorm_kernel<<<ROWS / 8, 256, 0, stream>>>(h, lnout_w, lnout_b, outp);
    mean_kernel<<<BATCH * DMODEL, 256, 0, stream>>>(outp, outp + (size_t)ROWS * DMODEL);
}